// ConvLayer_70763881169369
// MI455X (gfx1250) — compile-verified
//
#include <hip/hip_runtime.h>

#define N_NODES 50000
#define N_EDGES 500000
#define D 128
#define PS2 144  // pair-row stride in float2 units: 2*PS2 % 64 == 32 -> half-waves on disjoint bank halves

typedef __attribute__((ext_vector_type(2))) float v2f;
typedef __attribute__((ext_vector_type(8))) float v8f;

// ---------------------------------------------------------------------------
// Kernel 1: zero the workspace (agg[N*D] ++ cnt[N]) with 16B stores.
// ---------------------------------------------------------------------------
__global__ void zero_ws4(float4* __restrict__ ws, long n4) {
    long i = (long)blockIdx.x * blockDim.x + threadIdx.x;
    long stride = (long)gridDim.x * blockDim.x;
    float4 z = {0.0f, 0.0f, 0.0f, 0.0f};
    for (; i < n4; i += stride) ws[i] = z;
}

// ---------------------------------------------------------------------------
// Kernel 2: edge aggregation. One wave32 per edge: each lane moves 4 features
// (float4 gather from x[src], 4x global_atomic_add_f32 into agg[dst]).
// Lane 0 bumps the degree counter. x (25.6 MB) + agg (25.6 MB) stay L2-resident.
// ---------------------------------------------------------------------------
__global__ void aggregate(const float* __restrict__ x,
                          const long long* __restrict__ ei,
                          float* __restrict__ agg,
                          float* __restrict__ cnt) {
    long gid = (long)blockIdx.x * blockDim.x + threadIdx.x;
    long edge = gid >> 5;
    int lane = (int)(gid & 31);
    if (edge >= N_EDGES) return;

    long long src = ei[edge];
    long long dst = ei[N_EDGES + edge];

    const float4* xs = (const float4*)(x + (long)src * D);
    float4 v = xs[lane];

    float* a = agg + (long)dst * D + lane * 4;
    unsafeAtomicAdd(a + 0, v.x);
    unsafeAtomicAdd(a + 1, v.y);
    unsafeAtomicAdd(a + 2, v.z);
    unsafeAtomicAdd(a + 3, v.w);
    if (lane == 0) unsafeAtomicAdd(cnt + dst, 1.0f);
}

// ---------------------------------------------------------------------------
// Kernel 3: out = relu( (agg/max(cnt,1)) @ Wl + bl + x @ Wr )
// Full-precision f32 WMMA (V_WMMA_F32_16X16X4_F32), wave32, 16x16 tiles.
//   - Block = 256 threads = 8 waves; each wave owns one 16-row tile (all 128 cols).
//   - Wl/Wr staged in LDS in *pair-interleaved fragment layout*:
//       sW[p*PS2 + n] = { W[2p][n], W[2p+1][n] }   (float2, p = k/2)
//     so a B fragment is ONE ds_load_b64 per lane:
//       lanes 0-15  read pair p = k0/2     -> {W[k0][n],   W[k0+1][n]}
//       lanes 16-31 read pair p = k0/2 + 1 -> {W[k0+2][n], W[k0+3][n]}
//     PS2=144: the +1 pair row shifts banks by 32 -> conflict-free halves.
//   - A fragments (x / agg row pairs) are 8B global loads; mean formed in-reg
//     via precomputed 1/max(cnt,1).
// C/D layout: VGPR r -> M = r + (lane>=16 ? 8 : 0), N = lane&15.
// ---------------------------------------------------------------------------
__launch_bounds__(256)
__global__ void sage_gemm(const float* __restrict__ x,
                          const float* __restrict__ agg,
                          const float* __restrict__ cnt,
                          const float* __restrict__ Wl,
                          const float* __restrict__ bl,
                          const float* __restrict__ Wr,
                          float* __restrict__ out) {
    extern __shared__ v2f smem2[];
    v2f* sWl2 = smem2;                 // (D/2) pair-rows x PS2 stride
    v2f* sWr2 = smem2 + (D / 2) * PS2;

    // Cooperative staged load: build pair-interleaved layout.
    // Each iteration: float4 from row 2p and row 2p+1, emit 4 interleaved float2s.
    for (int idx4 = threadIdx.x; idx4 < (D / 2) * (D / 4); idx4 += 256) {
        int p  = idx4 >> 5;            // / (D/4)
        int n0 = (idx4 & 31) << 2;     // % (D/4) * 4
        float4 l0 = *(const float4*)(Wl + (2 * p) * D + n0);
        float4 l1 = *(const float4*)(Wl + (2 * p + 1) * D + n0);
        float4 r0 = *(const float4*)(Wr + (2 * p) * D + n0);
        float4 r1 = *(const float4*)(Wr + (2 * p + 1) * D + n0);
        v2f* dl = sWl2 + p * PS2 + n0;
        v2f* dr = sWr2 + p * PS2 + n0;
        dl[0] = (v2f){l0.x, l1.x}; dl[1] = (v2f){l0.y, l1.y};
        dl[2] = (v2f){l0.z, l1.z}; dl[3] = (v2f){l0.w, l1.w};
        dr[0] = (v2f){r0.x, r1.x}; dr[1] = (v2f){r0.y, r1.y};
        dr[2] = (v2f){r0.z, r1.z}; dr[3] = (v2f){r0.w, r1.w};
    }
    __syncthreads();

    int wave = threadIdx.x >> 5;
    int lane = threadIdx.x & 31;
    int tile = blockIdx.x * 8 + wave;          // 3125 total 16-row tiles
    if (tile >= N_NODES / 16) return;

    int row0 = tile * 16;
    int half = lane >> 4;                      // 0: lanes 0-15, 1: lanes 16-31
    int l16  = lane & 15;
    int arow = row0 + l16;                     // A-fragment row for this lane

    float c   = cnt[arow];
    float inv = 1.0f / fmaxf(c, 1.0f);

    const float* aX = x   + (long)arow * D + half * 2;
    const float* aM = agg + (long)arow * D + half * 2;

    v8f acc[8] = {};

    for (int k0 = 0; k0 < D; k0 += 4) {
        v2f fx = *(const v2f*)(aX + k0);       // x fragment
        v2f fm = *(const v2f*)(aM + k0);       // agg fragment
        fm *= inv;                             // -> mean fragment
        int p = (k0 >> 1) + half;              // pair row for this half-wave

#pragma unroll
        for (int t = 0; t < 8; ++t) {
            int n = t * 16 + l16;
            v2f bL = sWl2[p * PS2 + n];        // one ds_load_b64
            v2f bR = sWr2[p * PS2 + n];        // one ds_load_b64
            acc[t] = __builtin_amdgcn_wmma_f32_16x16x4_f32(
                false, fm, false, bL, (short)0, acc[t], false, false);
            acc[t] = __builtin_amdgcn_wmma_f32_16x16x4_f32(
                false, fx, false, bR, (short)0, acc[t], false, false);
        }
    }

    // Epilogue: bias + ReLU + store (each half-wave writes 64B contiguous runs).
    int rbase = row0 + half * 8;
#pragma unroll
    for (int t = 0; t < 8; ++t) {
        int colt = t * 16 + l16;
        float bias = bl[colt];
#pragma unroll
        for (int r = 0; r < 8; ++r) {
            float v = acc[t][r] + bias;
            out[(long)(rbase + r) * D + colt] = fmaxf(v, 0.0f);
        }
    }
}

// ---------------------------------------------------------------------------
extern "C" void kernel_launch(void* const* d_in, const int* in_sizes, int n_in,
                              void* d_out, int out_size, void* d_ws, size_t ws_size,
                              hipStream_t stream) {
    const float*     x  = (const float*)d_in[0];
    const long long* ei = (const long long*)d_in[1];   // int64 edge_index [2, E]
    const float*     Wl = (const float*)d_in[2];
    const float*     bl = (const float*)d_in[3];
    const float*     Wr = (const float*)d_in[4];
    float* out = (float*)d_out;

    float* agg = (float*)d_ws;                          // N*D floats (25.6 MB)
    float* cnt = agg + (size_t)N_NODES * D;             // N floats

    // 1) zero agg + cnt (re-init every call: deterministic)
    long n4 = ((long)N_NODES * D + N_NODES) / 4;        // divisible by 4
    zero_ws4<<<1024, 256, 0, stream>>>((float4*)d_ws, n4);

    // 2) edge aggregation: one wave per edge
    long nthreads = (long)N_EDGES * 32;
    int  blocks   = (int)((nthreads + 255) / 256);
    aggregate<<<blocks, 256, 0, stream>>>(x, ei, agg, cnt);

    // 3) fused mean-normalize + dual GEMM + bias + ReLU (f32 WMMA)
    int tiles   = N_NODES / 16;                         // 3125
    int gblocks = (tiles + 7) / 8;                      // 391
    size_t lds  = (size_t)2 * (D / 2) * PS2 * sizeof(v2f);  // 144 KB
    sage_gemm<<<gblocks, 256, lds, stream>>>(x, agg, cnt, Wl, bl, Wr, out);
}